// ESRNN_83150566850709
// MI455X (gfx1250) — compile-verified
//
#include <hip/hip_runtime.h>

typedef float v2f __attribute__((ext_vector_type(2)));
typedef float v8f __attribute__((ext_vector_type(8)));

#define B_    256
#define T_    512
#define IN_   48
#define H_    128
#define OUT_  48
#define SEAS_ 24
#define W_    465   // T - IN + 1
#define WO_   417   // T - OUT - IN + 1
#define TPAD_ 472   // W padded to multiple of 8 (max dilation)
#define SEASW_ 560  // T + SEAS + (OUT-SEAS)

__device__ __forceinline__ float sigm(float x) { return 1.f / (1.f + __expf(-x)); }

// ---------------- Exponential smoothing scan (sequential in T, parallel in B) ----
__global__ __launch_bounds__(256) void es_scan_kernel(
    const float* __restrict__ train,      // B x T
    const float* __restrict__ init_lev_sm,
    const float* __restrict__ init_seas_sm,
    const float* __restrict__ init_seas,  // SEAS
    float* __restrict__ levs,             // B x T
    float* __restrict__ seas)             // B x SEASW_
{
    __shared__ float ring[SEAS_ * B_];    // per-series rolling seasonality buffer
    const int b = threadIdx.x;
    const float lev_sm  = sigm(init_lev_sm[0]);
    const float seas_sm = sigm(init_seas_sm[0]);
    const float s00 = __expf(init_seas[0]);
#pragma unroll
    for (int j = 0; j < SEAS_; ++j) {
        seas[b * SEASW_ + j] = __expf(init_seas[j]);              // seas head [0..23]
        ring[j * B_ + b] = __expf(init_seas[(j + 1) % SEAS_]);    // buf0 = concat(s0[1:], s0[:1])
    }
    seas[b * SEASW_ + SEAS_] = s00;                               // seas head [24]
    float lev = train[b * T_] / s00;
    levs[b * T_] = lev;
    int pos = 0;
    for (int i = 1; i < T_; ++i) {
        const float xt = train[b * T_ + i];
        const float st = ring[pos * B_ + b];
        const float levn = lev_sm * (xt / st) + (1.f - lev_sm) * lev;
        const float sn = seas_sm * (xt / levn) + (1.f - seas_sm) * st;
        ring[pos * B_ + b] = sn;               // append at logical end
        pos = (pos + 1 == SEAS_) ? 0 : pos + 1;
        lev = levn;
        levs[b * T_ + i] = levn;
        seas[b * SEASW_ + SEAS_ + i] = sn;
    }
#pragma unroll
    for (int j = 0; j < SEAS_; ++j)           // seasonality extension (OUT > SEAS)
        seas[b * SEASW_ + 536 + j] = seas[b * SEASW_ + 512 + j];
}

// ---------------- Window builders ------------------------------------------------
__global__ void win_in_kernel(const float* __restrict__ train,
                              const float* __restrict__ levs,
                              const float* __restrict__ seas,
                              float* __restrict__ win)           // TPAD x B x IN
{
    const long idx = (long)blockIdx.x * 256 + threadIdx.x;
    if (idx >= (long)TPAD_ * B_ * IN_) return;
    const int k = (int)(idx % IN_);
    const int b = (int)((idx / IN_) % B_);
    const int t = (int)(idx / ((long)IN_ * B_));
    float v = 0.f;
    if (t < W_)
        v = train[b * T_ + t + k] / seas[b * SEASW_ + t + k] / levs[b * T_ + t + IN_ - 1];
    win[idx] = v;
}

__global__ void win_out_kernel(const float* __restrict__ train,
                               const float* __restrict__ levs,
                               const float* __restrict__ seas,
                               float* __restrict__ out)          // WO x B x OUT
{
    const long idx = (long)blockIdx.x * 256 + threadIdx.x;
    if (idx >= (long)WO_ * B_ * OUT_) return;
    const int k = (int)(idx % OUT_);
    const int b = (int)((idx / OUT_) % B_);
    const int t = (int)(idx / ((long)OUT_ * B_));
    out[idx] = train[b * T_ + t + IN_ + k] / seas[b * SEASW_ + t + IN_ + k]
             / levs[b * T_ + t + IN_ - 1];
}

// ---------------- Whole dilated-LSTM cell in ONE kernel --------------------------
// No cross-block dependency exists inside a cell: each block owns 16 folded-batch
// rows; its h/c recurrence is private, and step-i inputs come from the previous
// cell's completed buffer. So each block loops over all fold steps internally.
//   - h state lives in LDS (it is the WMMA A-fragment source), rewritten in place
//   - c state lives in registers (8 floats/lane == exact C/D fragment layout)
//   - wave w owns gate columns [16w,16w+16) across the 4 gate quarters (i,f,g,o)
template <int CIN, bool RES>
__global__ __launch_bounds__(256) void lstm_cell_kernel(
    const float* __restrict__ x,     // (steps*stride) x CIN input sequence rows
    const float* __restrict__ w_ih,  // 512 x CIN
    const float* __restrict__ w_hh,  // 512 x 128
    const float* __restrict__ b_ih,  // 512
    const float* __restrict__ b_hh,  // 512
    float* __restrict__ y,           // (steps*stride) x 128 output rows
    const float* __restrict__ res,   // residual rows (may alias y) or null
    int steps, int stride)           // stride = d*256 rows per fold step
{
    constexpr int LDX = CIN + 4;     // LDS pad -> conflict-free row reads
    constexpr int LDH = H_ + 4;
    __shared__ float sX[16 * LDX];
    __shared__ float sH[16 * LDH];

    const int tid = threadIdx.x;
    const int wave = tid >> 5;
    const int lane = tid & 31;
    const int lane_lo = lane & 15;
    const int lane_hi = lane >> 4;
    const int m0 = blockIdx.x << 4;

    const int ncol = (wave << 4) + lane_lo;  // column within gate quarter, 0..127
    const int koff = lane_hi << 1;           // K sub-offset per ISA A/B layout

    // zero-init hidden state (LDS) and cell state (registers)
    for (int i = tid; i < 16 * LDH; i += 256) sH[i] = 0.f;
    float creg[8];
#pragma unroll
    for (int e = 0; e < 8; ++e) creg[e] = 0.f;

    const float bI = b_ih[0 * H_ + ncol] + b_hh[0 * H_ + ncol];
    const float bF = b_ih[1 * H_ + ncol] + b_hh[1 * H_ + ncol];
    const float bG = b_ih[2 * H_ + ncol] + b_hh[2 * H_ + ncol];
    const float bO = b_ih[3 * H_ + ncol] + b_hh[3 * H_ + ncol];

    for (int s = 0; s < steps; ++s) {
        const size_t rowbase = (size_t)s * stride + m0;

        // stage this fold step's 16 input rows into LDS (shared by all 8 waves)
        for (int i = tid; i < 16 * CIN; i += 256) {
            const int r = i / CIN, col = i % CIN;
            sX[r * LDX + col] = x[(rowbase + r) * CIN + col];
        }
        if (s + 1 < steps)   // hint next step's rows toward the caches
            __builtin_prefetch(x + ((size_t)(s + 1) * stride + m0 + (tid >> 4)) * CIN, 0, 1);
        __syncthreads();     // sX ready; also orders prior sH writes vs reads below

        v8f acc[4];
#pragma unroll
        for (int g = 0; g < 4; ++g)
#pragma unroll
            for (int e = 0; e < 8; ++e) acc[g][e] = 0.f;

        // gates += x @ w_ih^T
        for (int kb = 0; kb < CIN; kb += 4) {
            v2f a;
            a.x = sX[lane_lo * LDX + kb + koff];
            a.y = sX[lane_lo * LDX + kb + koff + 1];
#pragma unroll
            for (int g = 0; g < 4; ++g) {
                const float* wp = w_ih + (size_t)(g * H_ + ncol) * CIN + kb + koff;
                v2f bf; bf.x = wp[0]; bf.y = wp[1];
                acc[g] = __builtin_amdgcn_wmma_f32_16x16x4_f32(
                    false, a, false, bf, (short)0, acc[g], false, false);
            }
        }
        // gates += h @ w_hh^T
        for (int kb = 0; kb < H_; kb += 4) {
            v2f a;
            a.x = sH[lane_lo * LDH + kb + koff];
            a.y = sH[lane_lo * LDH + kb + koff + 1];
#pragma unroll
            for (int g = 0; g < 4; ++g) {
                const float* wp = w_hh + (size_t)(g * H_ + ncol) * H_ + kb + koff;
                v2f bf; bf.x = wp[0]; bf.y = wp[1];
                acc[g] = __builtin_amdgcn_wmma_f32_16x16x4_f32(
                    false, a, false, bf, (short)0, acc[g], false, false);
            }
        }
        __syncthreads();     // all waves done reading sH before it is overwritten

        // fused LSTM nonlinearity; lane's 8 acc elements share one column ncol
#pragma unroll
        for (int e = 0; e < 8; ++e) {
            const int lr = (lane_hi << 3) + e;          // C/D layout: lanes 16-31 -> M+8
            const float ig = sigm(acc[0][e] + bI);
            const float fg = sigm(acc[1][e] + bF);
            const float gg = tanhf(acc[2][e] + bG);
            const float og = sigm(acc[3][e] + bO);
            const float cn = fg * creg[e] + ig * gg;
            const float hn = og * tanhf(cn);
            creg[e] = cn;
            sH[lr * LDH + ncol] = hn;                   // next step's A fragments
            const size_t off = (rowbase + lr) * H_ + ncol;
            float yv = hn;
            if (RES) yv += res[off];                    // read-then-write same addr: alias-safe
            y[off] = yv;
        }
    }
}

// ---------------- Dense head GEMM: out = act(A @ W^T + b) ------------------------
template <int N, int ACT>
__global__ __launch_bounds__(256) void gemm_kernel(
    const float* __restrict__ A,    // M x 128
    const float* __restrict__ Wt,   // N x 128
    const float* __restrict__ bias, // N
    float* __restrict__ out)        // M x N
{
    constexpr int LDA = H_ + 4;
    __shared__ float sA[16 * LDA];
    const int tid = threadIdx.x;
    const int wave = tid >> 5;
    const int lane = tid & 31;
    const int lane_lo = lane & 15;
    const int lane_hi = lane >> 4;
    const int m0 = blockIdx.x << 4;

    for (int i = tid; i < 16 * H_; i += 256) {
        const int r = i / H_, col = i % H_;
        sA[r * LDA + col] = A[(size_t)(m0 + r) * H_ + col];
    }
    __syncthreads();
    if ((wave << 4) >= N) return;

    const int ncol = (wave << 4) + lane_lo;
    const int koff = lane_hi << 1;
    v8f acc;
#pragma unroll
    for (int e = 0; e < 8; ++e) acc[e] = 0.f;

    for (int kb = 0; kb < H_; kb += 4) {
        v2f a;
        a.x = sA[lane_lo * LDA + kb + koff];
        a.y = sA[lane_lo * LDA + kb + koff + 1];
        const float* wp = Wt + (size_t)ncol * H_ + kb + koff;
        v2f bf; bf.x = wp[0]; bf.y = wp[1];
        acc = __builtin_amdgcn_wmma_f32_16x16x4_f32(
            false, a, false, bf, (short)0, acc, false, false);
    }
    const float bb = bias[ncol];
#pragma unroll
    for (int e = 0; e < 8; ++e) {
        const int row = m0 + (lane_hi << 3) + e;
        float v = acc[e] + bb;
        if (ACT) v = tanhf(v);
        out[(size_t)row * N + ncol] = v;
    }
}

// ---------------- Epilogue: holdout predictions ----------------------------------
__global__ void hold_kernel(const float* __restrict__ net_all,
                            const float* __restrict__ val,
                            const float* __restrict__ levs,
                            const float* __restrict__ seas,
                            float* __restrict__ hp, float* __restrict__ ha,
                            float* __restrict__ hn)
{
    const int i = blockIdx.x * 256 + threadIdx.x;
    if (i >= B_ * OUT_) return;
    const int b = i / OUT_, k = i % OUT_;
    const float lev = levs[b * T_ + T_ - 1];
    const float se = seas[b * SEASW_ + 512 + k];
    const float h = net_all[((size_t)(W_ - 1) * B_ + b) * OUT_ + k] * se * lev;
    hp[i] = h > 0.f ? h : 0.f;
    const float v = val[i];
    ha[i] = v;
    hn[i] = v / se / lev;
}

// ---------------- Host driver ----------------------------------------------------
extern "C" void kernel_launch(void* const* d_in, const int* in_sizes, int n_in,
                              void* d_out_v, int out_size, void* d_ws, size_t ws_size,
                              hipStream_t stream)
{
    const float* train  = (const float*)d_in[0];
    const float* val    = (const float*)d_in[1];
    const float* ilev   = (const float*)d_in[3];
    const float* iseas  = (const float*)d_in[4];
    const float* iseaso = (const float*)d_in[5];
    const float* w_ih[4]; const float* w_hh[4];
    const float* b_ih[4]; const float* b_hh[4];
    for (int i = 0; i < 4; ++i) {
        w_ih[i] = (const float*)d_in[6 + 4 * i];
        w_hh[i] = (const float*)d_in[7 + 4 * i];
        b_ih[i] = (const float*)d_in[8 + 4 * i];
        b_hh[i] = (const float*)d_in[9 + 4 * i];
    }
    const float* nl_w = (const float*)d_in[22];
    const float* nl_b = (const float*)d_in[23];
    const float* sc_w = (const float*)d_in[24];
    const float* sc_b = (const float*)d_in[25];

    float* out = (float*)d_out_v;
    float* ws  = (float*)d_ws;

    // workspace layout (floats); ~148 MB total -> entire working set is L2-resident
    float* levs = ws;                                        // 256*512
    float* seas = levs + (size_t)B_ * T_;                    // 256*560
    float* win  = seas + (size_t)B_ * SEASW_;                // 472*256*48
    float* bufA = win  + (size_t)TPAD_ * B_ * IN_;           // 472*256*128
    float* bufB = bufA + (size_t)TPAD_ * B_ * H_;            // 472*256*128

    // output layout (concatenated, return order)
    float* net_pred  = out;
    float* net_act   = net_pred  + (size_t)WO_ * B_ * OUT_;
    float* hold_pred = net_act   + (size_t)WO_ * B_ * OUT_;
    float* net_all   = hold_pred + (size_t)B_ * OUT_;
    float* hold_act  = net_all   + (size_t)W_ * B_ * OUT_;
    float* hold_norm = hold_act  + (size_t)B_ * OUT_;

    es_scan_kernel<<<1, 256, 0, stream>>>(train, ilev, iseas, iseaso, levs, seas);

    {
        const long n1 = (long)TPAD_ * B_ * IN_;
        win_in_kernel<<<(unsigned)((n1 + 255) / 256), 256, 0, stream>>>(train, levs, seas, win);
        const long n2 = (long)WO_ * B_ * OUT_;
        win_out_kernel<<<(unsigned)((n2 + 255) / 256), 256, 0, stream>>>(train, levs, seas, net_act);
    }

    // ResidualDRNN, one kernel per cell (fold steps looped inside the kernel)
    // cell 0: d=1, CIN=48, win -> bufA
    lstm_cell_kernel<IN_, false><<<256 / 16, 256, 0, stream>>>(
        win, w_ih[0], w_hh[0], b_ih[0], b_hh[0], bufA, nullptr, W_, 256);
    // cell 1: d=2, bufA -> bufB
    lstm_cell_kernel<H_, false><<<512 / 16, 256, 0, stream>>>(
        bufA, w_ih[1], w_hh[1], b_ih[1], b_hh[1], bufB, nullptr, (W_ + 1) / 2, 512);
    // cell 2: d=4, bufB -> bufA (bufB kept as residual)
    lstm_cell_kernel<H_, false><<<1024 / 16, 256, 0, stream>>>(
        bufB, w_ih[2], w_hh[2], b_ih[2], b_hh[2], bufA, nullptr, (W_ + 3) / 4, 1024);
    // cell 3: d=8, bufA -> bufB with residual += bufB (read-before-write, alias-safe)
    lstm_cell_kernel<H_, true><<<2048 / 16, 256, 0, stream>>>(
        bufA, w_ih[3], w_hh[3], b_ih[3], b_hh[3], bufB, bufB, (W_ + 7) / 8, 2048);

    // heads: z = tanh(x @ nl_w^T + nl_b) -> bufA (reuse); net_all = z @ sc_w^T + sc_b
    gemm_kernel<H_, 1><<<(W_ * B_) / 16, 256, 0, stream>>>(bufB, nl_w, nl_b, bufA);
    gemm_kernel<OUT_, 0><<<(W_ * B_) / 16, 256, 0, stream>>>(bufA, sc_w, sc_b, net_all);

    // network_pred is the causal prefix of net_out_all
    hipMemcpyAsync(net_pred, net_all, (size_t)WO_ * B_ * OUT_ * sizeof(float),
                   hipMemcpyDeviceToDevice, stream);

    hold_kernel<<<(B_ * OUT_ + 255) / 256, 256, 0, stream>>>(
        net_all, val, levs, seas, hold_pred, hold_act, hold_norm);
}